// TopKRouter_67284957659234
// MI455X (gfx1250) — compile-verified
//
#include <hip/hip_runtime.h>
#include <hip/hip_bf16.h>
#include <math.h>

typedef __attribute__((ext_vector_type(2))) float v2f;
typedef __attribute__((ext_vector_type(4))) float v4f;
typedef __attribute__((ext_vector_type(8))) float v8f;

#define D_MODEL   1024
#define N_EXPERT  64
#define TOPK      2
#define CAPACITY  1280
#define EBUF_CAP  8192      // per-expert candidate buffer (avg load 1024, huge margin)
#define TOK_BLK   64        // tokens per gemm block
#define KCHUNK    128       // K-tile staged in LDS
#define NCHUNK    (D_MODEL / KCHUNK)
#define NSORT     2048      // bitonic sort size (>= CAPACITY, power of 2)

// ---------------------------------------------------------------------------
// Phase 0: zero per-expert counters (must run every call: ws is not re-poisoned)
// ---------------------------------------------------------------------------
__global__ void init_cnt_kernel(int* cnt) {
    if (threadIdx.x < N_EXPERT) cnt[threadIdx.x] = 0;
}

// ---------------------------------------------------------------------------
// Phase 1: logits = x @ W^T + b   via V_WMMA_F32_16X16X4_F32 with async
//   global->LDS staging of x (GLOBAL_LOAD_ASYNC_TO_LDS_B128, ASYNCcnt-gated,
//   double-buffered so the next chunk fills while WMMAs run on this one).
//   block = 128 threads (4 waves). Block covers 64 tokens x 64 experts.
// ---------------------------------------------------------------------------
__device__ __forceinline__ void async_stage_chunk(
    const float* __restrict__ x, int t0, int kb, float* lds_base) {
    // 64 rows x 128 floats = 2048 float4s; 128 threads -> exactly 16 issues each
    const int tid = threadIdx.x;
    #pragma unroll
    for (int i = tid; i < (TOK_BLK * KCHUNK / 4); i += 128) {
        const int row = i >> 5;                 // 32 float4 per row
        const int c4  = (i & 31) << 2;
        const float* g = x + (size_t)(t0 + row) * D_MODEL + kb + c4;
        // LDS byte address = low 32 bits of the generic shared pointer
        const unsigned lds = (unsigned)(uintptr_t)(lds_base + row * KCHUNK + c4);
        asm volatile("global_load_async_to_lds_b128 %0, %1, off"
                     :: "v"(lds), "v"(g) : "memory");
    }
}

__global__ __launch_bounds__(128) void gemm_logits_kernel(
    const float* __restrict__ x, const float* __restrict__ W,
    const float* __restrict__ bias, float* __restrict__ logits) {
    __shared__ float xs[2][TOK_BLK][KCHUNK];   // 64 KB, double buffered

    const int tid  = threadIdx.x;
    const int wave = tid >> 5;                 // 0..3 -> expert tile
    const int lane = tid & 31;
    const int t0   = blockIdx.x * TOK_BLK;
    const int e0   = wave * 16;
    const int col  = lane & 15;                // A row (M) == B col (N) index
    const int koff = (lane >> 4) * 2;          // lanes 0-15: K,K+1 ; 16-31: K+2,K+3

    const v8f zero = {0.f,0.f,0.f,0.f,0.f,0.f,0.f,0.f};
    v8f acc[4];
    #pragma unroll
    for (int m = 0; m < 4; ++m) acc[m] = zero;

    const float* wrow_base = W + (size_t)(e0 + col) * D_MODEL;

    // prologue: kick off chunk 0
    async_stage_chunk(x, t0, 0, &xs[0][0][0]);

    for (int c = 0; c < NCHUNK; ++c) {
        // overlap: start filling the other buffer with chunk c+1
        if (c + 1 < NCHUNK) {
            async_stage_chunk(x, t0, (c + 1) * KCHUNK, &xs[(c + 1) & 1][0][0]);
            // async loads complete in order: <=16 outstanding => chunk c landed
            asm volatile("s_wait_asynccnt 0x10" ::: "memory");
        } else {
            asm volatile("s_wait_asynccnt 0x0" ::: "memory");
        }
        __syncthreads();                       // every wave's slice of chunk c in LDS

        const float* wrow = wrow_base + c * KCHUNK;
        const float (*xb)[KCHUNK] = xs[c & 1];
        for (int kk = 0; kk < KCHUNK; kk += 4) {
            // B (4x16) fragment: W[e0+col][kk+koff .. +1]
            const v2f bfrag = *(const v2f*)(wrow + kk + koff);
            #pragma unroll
            for (int m = 0; m < 4; ++m) {
                const v2f afrag = *(const v2f*)&xb[m * 16 + col][kk + koff];
                acc[m] = __builtin_amdgcn_wmma_f32_16x16x4_f32(
                    false, afrag, false, bfrag, (short)0, acc[m], false, false);
            }
        }
        __syncthreads();                       // done reading before buffer is refilled
    }

    // C/D layout: VGPR r -> (M=r, N=lane) for lanes 0-15, (M=r+8, N=lane-16) for 16-31
    const int mrow = (lane >> 4) * 8;
    const float bv = bias[e0 + col];
    #pragma unroll
    for (int m = 0; m < 4; ++m) {
        #pragma unroll
        for (int r = 0; r < 8; ++r) {
            const int trow = t0 + m * 16 + mrow + r;
            logits[(size_t)trow * N_EXPERT + e0 + col] = acc[m][r] + bv;
        }
    }
}

// ---------------------------------------------------------------------------
// Phase 2: fp32 softmax over 64 experts, top-2, atomic scatter to expert buffers
// ---------------------------------------------------------------------------
__global__ __launch_bounds__(256) void softmax_top2_kernel(
    const float* __restrict__ logits, float* __restrict__ probs,
    float* __restrict__ bufp, int* __restrict__ buft, int* __restrict__ cnt,
    int T) {
    const int t = blockIdx.x * 256 + threadIdx.x;
    if (t >= T) return;

    const float* row = logits + (size_t)t * N_EXPERT;
    float l[N_EXPERT];
    float mx = -INFINITY;
    #pragma unroll
    for (int e = 0; e < N_EXPERT; ++e) { l[e] = row[e]; mx = fmaxf(mx, l[e]); }
    float s = 0.f;
    #pragma unroll
    for (int e = 0; e < N_EXPERT; ++e) { l[e] = __expf(l[e] - mx); s += l[e]; }
    const float inv = 1.0f / s;

    float* prow = probs + (size_t)t * N_EXPERT;
    float p1 = -1.f, p2 = -1.f; int i1 = 0, i2 = 0;
    #pragma unroll
    for (int e = 0; e < N_EXPERT; ++e) {
        const float p = l[e] * inv;
        prow[e] = p;
        if (p > p1)      { p2 = p1; i2 = i1; p1 = p; i1 = e; }
        else if (p > p2) { p2 = p;  i2 = e; }
    }
    // routing slots in t*k + j order (top-1 then top-2)
    int pos = atomicAdd(&cnt[i1], 1);
    if (pos < EBUF_CAP) { bufp[(size_t)i1 * EBUF_CAP + pos] = p1;
                          buft[(size_t)i1 * EBUF_CAP + pos] = t; }
    pos = atomicAdd(&cnt[i2], 1);
    if (pos < EBUF_CAP) { bufp[(size_t)i2 * EBUF_CAP + pos] = p2;
                          buft[(size_t)i2 * EBUF_CAP + pos] = t; }
}

// ---------------------------------------------------------------------------
// Phase 3: per-expert capacity-limited top-1280, sorted descending.
//   Threshold via byte-radix histogram (only if n > 2048), then 2048-wide
//   LDS bitonic sort on positive-float bit patterns (uint order == float order).
// ---------------------------------------------------------------------------
__global__ __launch_bounds__(256) void expert_select_kernel(
    const float* __restrict__ bufp, const int* __restrict__ buft,
    const int* __restrict__ cnt, float* __restrict__ out_p,
    float* __restrict__ out_i) {
    __shared__ unsigned key[NSORT];
    __shared__ int      val[NSORT];
    __shared__ unsigned hist[256];
    __shared__ int      misc[2];

    const int e   = blockIdx.x;
    const int tid = threadIdx.x;
    int n = cnt[e]; if (n > EBUF_CAP) n = EBUF_CAP;
    const float* bp = bufp + (size_t)e * EBUF_CAP;
    const int*   bt = buft + (size_t)e * EBUF_CAP;

    for (int i = tid; i < NSORT; i += 256) { key[i] = 0u; val[i] = -1; }
    __syncthreads();

    if (n <= NSORT) {
        for (int i = tid; i < n; i += 256) {
            key[i] = __float_as_uint(bp[i]);
            val[i] = bt[i];
        }
    } else {
        // find bit-pattern of the CAPACITY-th largest prob
        unsigned chosen = 0u, maskhi = 0u;
        int remaining = CAPACITY;
        for (int shift = 24; shift >= 0; shift -= 8) {
            hist[tid] = 0u;
            __syncthreads();
            for (int i = tid; i < n; i += 256) {
                const unsigned kb = __float_as_uint(bp[i]);
                if ((kb & maskhi) == chosen)
                    atomicAdd(&hist[(kb >> shift) & 0xFFu], 1u);
            }
            __syncthreads();
            if (tid == 0) {
                int cum = 0, bin = 255;
                for (; bin > 0; --bin) {
                    const int c = (int)hist[bin];
                    if (cum + c >= remaining) break;
                    cum += c;
                }
                misc[0] = bin;
                misc[1] = remaining - cum;
            }
            __syncthreads();
            chosen   |= ((unsigned)misc[0]) << shift;
            remaining = misc[1];
            maskhi   |= 0xFFu << shift;
            __syncthreads();
        }
        if (tid == 0) { misc[0] = 0; misc[1] = 0; }
        __syncthreads();
        for (int i = tid; i < n; i += 256) {           // strictly above threshold
            const unsigned kb = __float_as_uint(bp[i]);
            if (kb > chosen) {
                const int p = atomicAdd(&misc[0], 1);
                if (p < NSORT) { key[p] = kb; val[p] = bt[i]; }
            }
        }
        __syncthreads();
        int ngt = misc[0]; if (ngt > NSORT) ngt = NSORT;
        for (int i = tid; i < n; i += 256) {           // fill ties at threshold
            const unsigned kb = __float_as_uint(bp[i]);
            if (kb == chosen) {
                const int p = atomicAdd(&misc[1], 1);
                const int slot = ngt + p;
                if (p < remaining && slot < NSORT) { key[slot] = kb; val[slot] = bt[i]; }
            }
        }
    }
    __syncthreads();

    // bitonic sort, descending (pads with key 0 sink to the end)
    for (int ks = 2; ks <= NSORT; ks <<= 1) {
        for (int j = ks >> 1; j > 0; j >>= 1) {
            for (int i = tid; i < NSORT; i += 256) {
                const int ixj = i ^ j;
                if (ixj > i) {
                    const bool desc = ((i & ks) == 0);
                    const unsigned ka = key[i], kb = key[ixj];
                    if ((ka < kb) == desc) {
                        key[i] = kb; key[ixj] = ka;
                        const int va = val[i]; val[i] = val[ixj]; val[ixj] = va;
                    }
                }
            }
            __syncthreads();
        }
    }

    for (int i = tid; i < CAPACITY; i += 256) {
        out_p[(size_t)e * CAPACITY + i] = __uint_as_float(key[i]);  // 0.0f padding
        out_i[(size_t)e * CAPACITY + i] = (float)val[i];            // -1 padding
    }
}

// ---------------------------------------------------------------------------
extern "C" void kernel_launch(void* const* d_in, const int* in_sizes, int n_in,
                              void* d_out, int out_size, void* d_ws, size_t ws_size,
                              hipStream_t stream) {
    const float* x    = (const float*)d_in[0];
    const float* W    = (const float*)d_in[1];
    const float* bias = (const float*)d_in[2];
    (void)n_in; (void)out_size; (void)ws_size;   // k=2, capacity=1280 fixed by problem

    const int T = in_sizes[0] / D_MODEL;         // 32768

    float* logits = (float*)d_out;                               // (T, 64)
    float* probs  = logits + (size_t)T * N_EXPERT;               // (T, 64)
    float* out_p  = probs  + (size_t)T * N_EXPERT;               // (64, 1280)
    float* out_i  = out_p  + (size_t)N_EXPERT * CAPACITY;        // (64, 1280)

    char* ws   = (char*)d_ws;
    int*  cnt  = (int*)ws;                                                   // 256 B
    float* bufp = (float*)(ws + 256);                                        // 2 MB
    int*   buft = (int*)(ws + 256 + (size_t)N_EXPERT * EBUF_CAP * sizeof(float)); // 2 MB

    init_cnt_kernel<<<1, 64, 0, stream>>>(cnt);
    gemm_logits_kernel<<<T / TOK_BLK, 128, 0, stream>>>(x, W, bias, logits);
    softmax_top2_kernel<<<(T + 255) / 256, 256, 0, stream>>>(logits, probs,
                                                             bufp, buft, cnt, T);
    expert_select_kernel<<<N_EXPERT, 256, 0, stream>>>(bufp, buft, cnt, out_p, out_i);
}